// TripletLoss_73504070303903
// MI455X (gfx1250) — compile-verified
//
#include <hip/hip_runtime.h>

#define BB 4096
#define NN 8192
#define DD 512
#define MARGIN_F 1.0f

typedef float v2f __attribute__((ext_vector_type(2)));
typedef float v8f __attribute__((ext_vector_type(8)));

// ---------------------------------------------------------------------------
// Kernel 0: zero the scalar accumulator
// ---------------------------------------------------------------------------
__global__ void init_kernel(float* __restrict__ acc) {
  if (threadIdx.x == 0 && blockIdx.x == 0) acc[0] = 0.0f;
}

// ---------------------------------------------------------------------------
// Kernel 1: per anchor row, find the one-hot positive index and compute
// pos_dist[b] = 1 - dot(A[b], S[j]).  One block per row.
// ---------------------------------------------------------------------------
__global__ void __launch_bounds__(256)
posdist_kernel(const float* __restrict__ A, const float* __restrict__ S,
               const long long* __restrict__ P, float* __restrict__ posd) {
  const int b   = blockIdx.x;
  const int tid = threadIdx.x;

  __shared__ int sj;
  if (tid == 0) sj = 0;
  __syncthreads();

  const long long* row = P + (size_t)b * NN;
  for (int n = tid; n < NN; n += 256) {
    if (row[n] != 0) sj = n;  // exactly one nonzero per row -> no race
  }
  __syncthreads();
  const int j = sj;

  const float* a = A + (size_t)b * DD;
  const float* s = S + (size_t)j * DD;
  float part = a[tid] * s[tid] + a[tid + 256] * s[tid + 256];

  __shared__ float red[256];
  red[tid] = part;
  __syncthreads();
  for (int off = 128; off > 0; off >>= 1) {
    if (tid < off) red[tid] += red[tid + off];
    __syncthreads();
  }
  if (tid == 0) posd[b] = 1.0f - red[0];
}

// ---------------------------------------------------------------------------
// Fragment bundle for one K-step (K=4): two A tiles (32 rows) x four B tiles
// (64 cols).  f32 WMMA 16x16x4 layout: lanes 0-15 hold K={k,k+1}, lanes 16-31
// hold K={k+2,k+3}; the +kh is folded into the base pointers.
// ---------------------------------------------------------------------------
struct Frag {
  v2f a0, a1, b0, b1, b2, b3;
};

__device__ __forceinline__ Frag
load_frag(const float* __restrict__ A0, const float* __restrict__ A1,
          const float* __restrict__ B0, const float* __restrict__ B1,
          const float* __restrict__ B2, const float* __restrict__ B3, int k) {
  Frag f;
  f.a0 = *(const v2f*)(A0 + k);
  f.a1 = *(const v2f*)(A1 + k);
  f.b0 = *(const v2f*)(B0 + k);
  f.b1 = *(const v2f*)(B1 + k);
  f.b2 = *(const v2f*)(B2 + k);
  f.b3 = *(const v2f*)(B3 + k);
  return f;
}

#define WMMA_F32(a, b, c) \
  __builtin_amdgcn_wmma_f32_16x16x4_f32(false, (a), false, (b), (short)0, (c), false, false)

__device__ __forceinline__ void
mma_step(const Frag& f, v8f& c00, v8f& c01, v8f& c02, v8f& c03,
                        v8f& c10, v8f& c11, v8f& c12, v8f& c13) {
  c00 = WMMA_F32(f.a0, f.b0, c00);
  c01 = WMMA_F32(f.a0, f.b1, c01);
  c02 = WMMA_F32(f.a0, f.b2, c02);
  c03 = WMMA_F32(f.a0, f.b3, c03);
  c10 = WMMA_F32(f.a1, f.b0, c10);
  c11 = WMMA_F32(f.a1, f.b1, c11);
  c12 = WMMA_F32(f.a1, f.b2, c12);
  c13 = WMMA_F32(f.a1, f.b3, c13);
}

// ---------------------------------------------------------------------------
// Kernel 2: fused WMMA GEMM + hinge + reduction.
// Block = 256 threads = 8 waves (4 M-waves x 2 N-waves); each wave computes a
// 32(M) x 64(N) strip (8 output tiles); block tile is 128M x 128N.
// Two-stage software pipeline, and each stage owns a PRIVATE accumulator set
// (cA / cB, summed in the epilogue) so every wmma chain ties D==C in fixed
// registers -- no cross-stage accumulator copies.
// hinge(b,n) = max(pos_dist[b] + dot(b,n), 0)  (MARGIN=1 folded in; the
// positive column's contribution, exactly MARGIN, is removed in finalize).
// ---------------------------------------------------------------------------
__global__ void __launch_bounds__(256)
gemm_hinge_kernel(const float* __restrict__ A, const float* __restrict__ S,
                  const float* __restrict__ posd, float* __restrict__ acc) {
  const int lane  = threadIdx.x & 31;
  const int wid   = threadIdx.x >> 5;
  const int lm    = lane & 15;   // M (for A) / N (for B) within a tile
  const int hi    = lane >> 4;   // lane half selects K sub-range
  const int kh    = hi * 2;
  const int waveN = wid & 1;
  const int waveM = wid >> 1;
  const int m0 = blockIdx.y * 128 + waveM * 32;
  const int n0 = blockIdx.x * 128 + waveN * 64;

  const float* A0 = A + (size_t)(m0 +  0 + lm) * DD + kh;
  const float* A1 = A + (size_t)(m0 + 16 + lm) * DD + kh;
  const float* B0 = S + (size_t)(n0 +  0 + lm) * DD + kh;
  const float* B1 = S + (size_t)(n0 + 16 + lm) * DD + kh;
  const float* B2 = S + (size_t)(n0 + 32 + lm) * DD + kh;
  const float* B3 = S + (size_t)(n0 + 48 + lm) * DD + kh;

  // stage-0 accumulators
  v8f a00 = {}, a01 = {}, a02 = {}, a03 = {};
  v8f a10 = {}, a11 = {}, a12 = {}, a13 = {};
  // stage-1 accumulators
  v8f b00 = {}, b01 = {}, b02 = {}, b03 = {};
  v8f b10 = {}, b11 = {}, b12 = {}, b13 = {};

  Frag p0 = load_frag(A0, A1, B0, B1, B2, B3, 0);
  Frag p1 = load_frag(A0, A1, B0, B1, B2, B3, 4);
  for (int k = 8; k < DD; k += 8) {
    mma_step(p0, a00, a01, a02, a03, a10, a11, a12, a13);
    p0 = load_frag(A0, A1, B0, B1, B2, B3, k);
    mma_step(p1, b00, b01, b02, b03, b10, b11, b12, b13);
    p1 = load_frag(A0, A1, B0, B1, B2, B3, k + 4);
  }
  mma_step(p0, a00, a01, a02, a03, a10, a11, a12, a13);
  mma_step(p1, b00, b01, b02, b03, b10, b11, b12, b13);

  // merge the two stage accumulators
  v8f c00 = a00 + b00, c01 = a01 + b01, c02 = a02 + b02, c03 = a03 + b03;
  v8f c10 = a10 + b10, c11 = a11 + b11, c12 = a12 + b12, c13 = a13 + b13;

  // C/D layout: VGPR v of tile (mi,t) holds (M = m0 + mi*16 + v + hi*8,
  //                                          N = n0 + t*16 + lm)
  float pd0[8], pd1[8];
#pragma unroll
  for (int v = 0; v < 8; ++v) {
    pd0[v] = posd[m0 + hi * 8 + v];
    pd1[v] = posd[m0 + 16 + hi * 8 + v];
  }

  float s = 0.0f;
#pragma unroll
  for (int v = 0; v < 8; ++v) {
    s += fmaxf(pd0[v] + c00[v], 0.0f);
    s += fmaxf(pd0[v] + c01[v], 0.0f);
    s += fmaxf(pd0[v] + c02[v], 0.0f);
    s += fmaxf(pd0[v] + c03[v], 0.0f);
    s += fmaxf(pd1[v] + c10[v], 0.0f);
    s += fmaxf(pd1[v] + c11[v], 0.0f);
    s += fmaxf(pd1[v] + c12[v], 0.0f);
    s += fmaxf(pd1[v] + c13[v], 0.0f);
  }

  __shared__ float red[256];
  red[threadIdx.x] = s;
  __syncthreads();
  for (int off = 128; off > 0; off >>= 1) {
    if ((int)threadIdx.x < off) red[threadIdx.x] += red[threadIdx.x + off];
    __syncthreads();
  }
  if (threadIdx.x == 0) atomicAdd(acc, red[0]);
}

// ---------------------------------------------------------------------------
// Kernel 3: remove the positive columns' MARGIN contribution and normalize.
// ---------------------------------------------------------------------------
__global__ void finalize_kernel(const float* __restrict__ acc, float* __restrict__ out) {
  if (threadIdx.x == 0 && blockIdx.x == 0) {
    out[0] = (acc[0] - (float)BB * MARGIN_F) / ((float)(NN - 1) * (float)BB);
  }
}

extern "C" void kernel_launch(void* const* d_in, const int* in_sizes, int n_in,
                              void* d_out, int out_size, void* d_ws, size_t ws_size,
                              hipStream_t stream) {
  (void)in_sizes; (void)n_in; (void)out_size; (void)ws_size;
  const float*     A = (const float*)d_in[0];      // [BB, DD]
  const float*     S = (const float*)d_in[1];      // [NN, DD]
  const long long* P = (const long long*)d_in[2];  // [BB, NN] one-hot int64
  float* out = (float*)d_out;

  float* ws   = (float*)d_ws;
  float* acc  = ws;        // 1 float (padded to 8)
  float* posd = ws + 8;    // BB floats

  init_kernel<<<1, 32, 0, stream>>>(acc);
  posdist_kernel<<<BB, 256, 0, stream>>>(A, S, P, posd);
  dim3 grid(NN / 128, BB / 128);
  gemm_hinge_kernel<<<grid, 256, 0, stream>>>(A, S, posd, acc);
  finalize_kernel<<<1, 32, 0, stream>>>(acc, out);
}